// NemotronHTopkRouter_2216203125391
// MI455X (gfx1250) — compile-verified
//
#include <hip/hip_runtime.h>
#include <math.h>

typedef __attribute__((ext_vector_type(2))) float v2f;
typedef __attribute__((ext_vector_type(4))) float v4f;
typedef __attribute__((ext_vector_type(8))) float v8f;

#define H_DIM      4096
#define N_EXPERTS  256
#define TOPK       8
#define NGROUP     8
#define TOPK_GROUP 4
#define RSCALE     2.5f

#define TM       64                  // tokens per block (4 tiles of 16)
#define MT       4                   // token tiles per wave
#define CHUNK_K  128
#define LDA      (CHUNK_K + 4)       // padded row stride: kills 16-way bank conflicts
#define NCHUNK   (H_DIM / CHUNK_K)   // 32
#define WSTEPS   (CHUNK_K / 4)       // 32 B fragments / chunk
#define NTHREADS 512                 // 16 wave32s

// A ping-pong (2*64*132 = 16896 f) unioned with logits (64*256 = 16384 f)
#define SMEM_F   (2 * TM * LDA)

__global__ __launch_bounds__(NTHREADS)
void nemotron_router_kernel(const float* __restrict__ hidden,   // [T, H]
                            const float* __restrict__ weight,   // [E, H]
                            const float* __restrict__ bias,     // [E]
                            float* __restrict__ out,            // [T*8 idx][T*8 w]
                            int T) {
    __shared__ float smem[SMEM_F];                 // ~66 KB, dual-purpose

    const int t0   = blockIdx.x * TM;
    const int tid  = threadIdx.x;
    const int wave = tid >> 5;                     // 0..15
    const int lane = tid & 31;
    const int half = lane >> 4;                    // K sub-block select (0/1)
    const int l16  = lane & 15;

    // ---------------- Phase 1: fp32 WMMA GEMM ------------------------------
    // wave w: experts [16w,16w+16) x 4 token tiles (rows 16m..16m+15)
    const int e0 = wave * 16;
    // B lane layout 4x16 fp32: v0=K{0|2}, v1=K{1|3}; N = lane&15
    const float* wrow = weight + (size_t)(e0 + l16) * H_DIM + 2 * half;
    // A staging: 64 rows x 128 cols per chunk; 8 threads/row x 16 floats
    const int srow = tid >> 3;                     // 0..63
    const int scol = (tid & 7) * 16;               // 0..112
    const float* arow = hidden + (size_t)(t0 + srow) * H_DIM + scol;

    v8f acc[MT] = {};
    v2f bcur[WSTEPS];

    // ---- prologue: stage A chunk 0, B chunk 0 to registers ----
    {
        const int sb = srow * LDA + scol;
        #pragma unroll
        for (int p = 0; p < 4; ++p)
            *(v4f*)&smem[sb + 4 * p] = *(const v4f*)(arow + 4 * p);
        #pragma unroll
        for (int i = 0; i < WSTEPS; ++i)
            bcur[i] = *(const v2f*)(wrow + 4 * i);
    }
    __syncthreads();

    // ---- main loop: compute chunk c, prefetch chunk c+1 ----
    for (int c = 0; c < NCHUNK - 1; ++c) {
        const int kn   = (c + 1) * CHUNK_K;
        const int base = (c & 1) * (TM * LDA);
        v4f an[4];
        #pragma unroll
        for (int p = 0; p < 4; ++p)
            an[p] = *(const v4f*)(arow + kn + 4 * p);   // next A, hidden by WMMAs
        const float* wpn = wrow + kn;

        #pragma unroll
        for (int i = 0; i < WSTEPS; ++i) {
            v2f b = bcur[i];
            #pragma unroll
            for (int m = 0; m < MT; ++m) {
                // A lane layout 16x4 fp32: M = lane&15; v0=K{0|2}, v1=K{1|3}
                v2f a = *(const v2f*)&smem[base + (16 * m + l16) * LDA + 4 * i + 2 * half];
                acc[m] = __builtin_amdgcn_wmma_f32_16x16x4_f32(
                             false, a, false, b, (short)0, acc[m], false, false);
            }
            bcur[i] = *(const v2f*)(wpn + 4 * i);       // rolling B prefetch
        }
        const int nb = ((c & 1) ^ 1) * (TM * LDA) + srow * LDA + scol;
        #pragma unroll
        for (int p = 0; p < 4; ++p)
            *(v4f*)&smem[nb + 4 * p] = an[p];
        __syncthreads();
    }
    // ---- epilogue: last chunk, compute only ----
    {
        const int base = ((NCHUNK - 1) & 1) * (TM * LDA);
        #pragma unroll
        for (int i = 0; i < WSTEPS; ++i) {
            v2f b = bcur[i];
            #pragma unroll
            for (int m = 0; m < MT; ++m) {
                v2f a = *(const v2f*)&smem[base + (16 * m + l16) * LDA + 4 * i + 2 * half];
                acc[m] = __builtin_amdgcn_wmma_f32_16x16x4_f32(
                             false, a, false, b, (short)0, acc[m], false, false);
            }
        }
    }
    __syncthreads();   // all A reads done; smem now becomes the logits buffer

    // C layout 16x16 fp32: lane L, vgpr j -> M = j + 8*(L>>4), N = L&15
    #pragma unroll
    for (int m = 0; m < MT; ++m)
        #pragma unroll
        for (int j = 0; j < 8; ++j)
            smem[(16 * m + j + 8 * half) * N_EXPERTS + e0 + l16] = acc[m][j];
    __syncthreads();

    // ---------------- Phase 2: grouped top-k, wave w owns tokens 4w..4w+3 --
    const int base_e = lane * 8;               // 8 contiguous experts per lane
    float bs[8];
    #pragma unroll
    for (int j = 0; j < 8; ++j) bs[j] = bias[base_e + j];   // hoisted

    for (int q = 0; q < MT; ++q) {
        const int tl = wave * MT + q;          // local token
        const float* lrow = &smem[tl * N_EXPERTS];

        float cs[8];
        #pragma unroll
        for (int j = 0; j < 8; ++j) {
            float s = 1.0f / (1.0f + __expf(-lrow[base_e + j]));
            cs[j] = s + bs[j];                 // corrected: for CHOICE only
        }

        // lane-local top-2 of corrected scores
        float m1 = -INFINITY, m2 = -INFINITY;
        #pragma unroll
        for (int j = 0; j < 8; ++j) {
            if (cs[j] > m1) { m2 = m1; m1 = cs[j]; }
            else if (cs[j] > m2) { m2 = cs[j]; }
        }
        // merge across the 4 lanes of each 32-expert group
        #pragma unroll
        for (int off = 1; off <= 2; off <<= 1) {
            float o1 = __shfl_xor(m1, off, 32);
            float o2 = __shfl_xor(m2, off, 32);
            float n1 = fmaxf(m1, o1);
            float n2 = fmaxf(fminf(m1, o1), fmaxf(m2, o2));
            m1 = n1; m2 = n2;
        }
        float group_score = m1 + m2;

        // rank my group among 8 (ties -> lower index first)
        float gs[NGROUP];
        #pragma unroll
        for (int g = 0; g < NGROUP; ++g) gs[g] = __shfl(group_score, g * 4, 32);
        const int myg = lane >> 2;
        int rank = 0;
        #pragma unroll
        for (int g = 0; g < NGROUP; ++g) {
            if (g == myg) continue;
            if (gs[g] > gs[myg] || (gs[g] == gs[myg] && g < myg)) rank++;
        }
        const bool kept = (rank < TOPK_GROUP);
        #pragma unroll
        for (int j = 0; j < 8; ++j) cs[j] = kept ? cs[j] : 0.0f;

        // iterative wave-wide top-8 (value desc, index asc on ties)
        float topw[TOPK];
        int   topi[TOPK];
        float wsum = 1e-20f;
        for (int s = 0; s < TOPK; ++s) {
            float bv = cs[0]; int bj = 0;
            #pragma unroll
            for (int j = 1; j < 8; ++j)
                if (cs[j] > bv) { bv = cs[j]; bj = j; }
            float v = bv; int idx = base_e + bj;
            #pragma unroll
            for (int off = 16; off > 0; off >>= 1) {
                float ov = __shfl_xor(v, off, 32);
                int   oi = __shfl_xor(idx, off, 32);
                if (ov > v || (ov == v && oi < idx)) { v = ov; idx = oi; }
            }
            topi[s] = idx;
            // weight from UN-corrected sigmoid (recompute from logit)
            float w = 1.0f / (1.0f + __expf(-lrow[idx]));   // LDS broadcast
            topw[s] = w;
            wsum += w;
            if (idx >= base_e && idx < base_e + 8) cs[idx - base_e] = -INFINITY;
        }
        const float scale = RSCALE / wsum;

        if (lane == 0) {
            const int tg = t0 + tl;
            float* outI = out;                      // indices (as float values)
            float* outW = out + (size_t)T * TOPK;   // normalized weights
            #pragma unroll
            for (int s = 0; s < TOPK; ++s) {
                outI[(size_t)tg * TOPK + s] = (float)topi[s];
                outW[(size_t)tg * TOPK + s] = topw[s] * scale;
            }
        }
    }
}

extern "C" void kernel_launch(void* const* d_in, const int* in_sizes, int n_in,
                              void* d_out, int out_size, void* d_ws, size_t ws_size,
                              hipStream_t stream) {
    (void)n_in; (void)out_size; (void)d_ws; (void)ws_size;
    const float* hidden = (const float*)d_in[0];
    const float* weight = (const float*)d_in[1];
    const float* bias   = (const float*)d_in[2];
    float* out = (float*)d_out;
    const int T = in_sizes[0] / H_DIM;          // 8192
    dim3 grid(T / TM), block(NTHREADS);
    nemotron_router_kernel<<<grid, block, 0, stream>>>(hidden, weight, bias, out, T);
}